// Ablock_70712341561650
// MI455X (gfx1250) — compile-verified
//
#include <hip/hip_runtime.h>
#include <cstdint>

// Ablock spatial paradigm, fused single pass.
//   x: (8, 320, 128, 128) fp32.  Per channel c, _lem is a circular 1-px shift
//   whose direction depends on c%5; lem0 and lem1 shift the SAME direction,
//   so out(i,j) needs x at shift-0, shift-1, shift-2 only (wrap mod 128).
// Memory-bound: ~336 MB min traffic -> ~14.4 us @ 23.3 TB/s.
// CDNA5 path: async global->LDS staging (ASYNCcnt) + NT b128 stores.

#define HW_PLANE 16384   // 128*128
#define C_ 320

// Native vector types (HIP's float4 is a class; builtins need real vectors)
typedef int   v4i __attribute__((vector_size(16)));
typedef float v4f __attribute__((ext_vector_type(4)));

typedef __attribute__((address_space(1))) v4i as1_v4i;
typedef __attribute__((address_space(3))) v4i as3_v4i;

__global__ __launch_bounds__(256) void ablock_kernel(const float* __restrict__ x,
                                                     float* __restrict__ out) {
    __shared__ __align__(16) float tile[HW_PLANE];   // 64 KB: one (b,c) plane

    const int plane = blockIdx.x;
    const int c = plane % C_;
    const int r = c % 5;
    // shift direction for this channel (uniform across block)
    int di = 0, dj = 0;
    if      (r == 0) dj =  1;   // out(i,j) = x(i, j+1)
    else if (r == 1) dj = -1;   // out(i,j) = x(i, j-1)
    else if (r == 2) di =  1;   // out(i,j) = x(i+1, j)
    else if (r == 3) di = -1;   // out(i,j) = x(i-1, j)
    // r == 4: identity

    const float* src = x + (size_t)plane * HW_PLANE;

    // ---- Stage full plane into LDS via async global->LDS b128 copies ----
#pragma unroll
    for (int k = 0; k < 16; ++k) {
        const int v = (k << 8) + threadIdx.x;          // float4 index [0,4096)
#if defined(__HIP_DEVICE_COMPILE__) && __has_builtin(__builtin_amdgcn_global_load_async_to_lds_b128)
        __builtin_amdgcn_global_load_async_to_lds_b128(
            (as1_v4i*)(uintptr_t)(src + (size_t)v * 4),
            (as3_v4i*)(uint32_t)(uintptr_t)(&tile[v * 4]),
            0, 0);
#else
#if defined(__HIP_DEVICE_COMPILE__)
#warning "gfx1250 async-to-LDS builtin unavailable; using synchronous fallback"
#endif
        ((v4f*)tile)[v] = ((const v4f*)src)[v];
#endif
    }

#if defined(__HIP_DEVICE_COMPILE__)
#if __has_builtin(__builtin_amdgcn_s_wait_asynccnt)
    __builtin_amdgcn_s_wait_asynccnt(0);
#else
    asm volatile("s_wait_asynccnt 0" ::: "memory");
#endif
#endif
    __syncthreads();

    // ---- Compute 64 outputs per thread (16x float4), NT b128 stores ----
    float* dst = out + (size_t)plane * HW_PLANE;
    for (int k = 0; k < 16; ++k) {
        const int v  = (k << 8) + threadIdx.x;   // float4 index
        const int e  = v << 2;                   // element index
        const int i  = e >> 7;                   // row
        const int j0 = e & 127;                  // col of lane's first element
        v4f o;
#pragma unroll
        for (int q = 0; q < 4; ++q) {
            const int j  = j0 + q;               // j0 multiple of 4 -> j < 128
            const float a  = tile[e + q];                                              // x
            const float b  = tile[(((i +     di) & 127) << 7) | ((j +     dj) & 127)]; // shift 1
            const float c2 = tile[(((i + 2 * di) & 127) << 7) | ((j + 2 * dj) & 127)]; // shift 2
            const float f    = b / (1.0f + __expf(-b));                 // silu(lem0 x)
            const float sc2  = c2 / (1.0f + __expf(-c2));               // silu at double shift
            const float fatt = 1.0f / (1.0f + __expf(-sc2)) - 0.5f;     // sigmoid(f)@shift - 0.5
            o[q] = (f + a) * fatt;
        }
        __builtin_nontemporal_store(o, (v4f*)dst + v);
    }
}

extern "C" void kernel_launch(void* const* d_in, const int* in_sizes, int n_in,
                              void* d_out, int out_size, void* d_ws, size_t ws_size,
                              hipStream_t stream) {
    const float* x = (const float*)d_in[0];
    float* out = (float*)d_out;
    const int planes = in_sizes[0] / HW_PLANE;    // 8*320 = 2560
    ablock_kernel<<<dim3(planes), dim3(256), 0, stream>>>(x, out);
}